// EdgeFeature_79809082295189
// MI455X (gfx1250) — compile-verified
//
#include <hip/hip_runtime.h>
#include <hip/hip_bf16.h>

typedef __attribute__((ext_vector_type(2))) float v2f;
typedef __attribute__((ext_vector_type(8))) float v8f;

#define C_DIM 64
#define N_PTS 2048
#define K_NN  16
#define B_SZ  8

// -------- squared norms: sq[b*N + n] = sum_c x[b,c,n]^2 --------
__global__ void __launch_bounds__(256)
norms_kernel(const float* __restrict__ x, float* __restrict__ sq) {
    int idx = blockIdx.x * 256 + threadIdx.x;
    int b = idx / N_PTS, n = idx % N_PTS;
    const float* X = x + (size_t)b * C_DIM * N_PTS;
    float s = 0.f;
    #pragma unroll
    for (int c = 0; c < C_DIM; ++c) {
        float v = X[c * N_PTS + n];
        s += v * v;
    }
    sq[idx] = s;
}

// -------- distance matrix via WMMA f32 16x16x4 --------
// One wave computes one 16x16 tile of D[b] = sq[n] + sq[m] - 2 * X^T X.
// grid = (16, 128, b_count), block = 256 (8 waves; wave -> m-tile)
__global__ void __launch_bounds__(256)
dist_wmma_kernel(const float* __restrict__ x, const float* __restrict__ sq,
                 float* __restrict__ D, int b_start, unsigned long long d_batch_stride) {
    int lane = threadIdx.x & 31;
    int wave = threadIdx.x >> 5;
    int bl   = blockIdx.z;
    int b    = b_start + bl;
    int n0   = blockIdx.y * 16;
    int m0   = (blockIdx.x * 8 + wave) * 16;

    const float* X = x + (size_t)b * C_DIM * N_PTS;
    int half = lane >> 4;        // 0: lanes 0-15, 1: lanes 16-31
    int l15  = lane & 15;
    int koff = half * 2;         // VGPR0 holds K=0 (lo lanes) / K=2 (hi lanes)

    const float* Acol = X + n0 + l15;   // A row index = n, K = channel
    const float* Bcol = X + m0 + l15;   // B col index = m, K = channel

    v8f acc = {};
    #pragma unroll
    for (int kk = 0; kk < 16; ++kk) {
        int c0 = kk * 4 + koff;
        v2f a, bm;
        a.x  = Acol[(c0 + 0) * N_PTS];
        a.y  = Acol[(c0 + 1) * N_PTS];
        bm.x = Bcol[(c0 + 0) * N_PTS];
        bm.y = Bcol[(c0 + 1) * N_PTS];
        acc = __builtin_amdgcn_wmma_f32_16x16x4_f32(
            /*neg_a=*/false, a, /*neg_b=*/false, bm,
            /*c_mod=*/(short)0, acc, /*reuse_a=*/false, /*reuse_b=*/false);
    }

    const float* sqb = sq + (size_t)b * N_PTS;
    float sm = sqb[m0 + l15];
    float* Db = D + (size_t)bl * d_batch_stride;
    #pragma unroll
    for (int r = 0; r < 8; ++r) {
        int n = n0 + r + half * 8;           // C tile: VGPR r holds rows r, r+8
        float d = sqb[n] + sm - 2.0f * acc[r];
        Db[(size_t)n * N_PTS + (m0 + l15)] = d;
    }
}

// -------- per-row top-16 (excluding self), ascending, stable ties --------
// Reads D column-wise (symmetric matrix -> coalesced across threads).
// grid = (N/256, b_count), block = 256. One thread per row n.
__global__ void __launch_bounds__(256)
topk_kernel(const float* __restrict__ D, int* __restrict__ idxbuf,
            float* __restrict__ idx_out, int b_start,
            unsigned long long d_batch_stride, unsigned long long idx_batch_stride) {
    int n  = blockIdx.x * 256 + threadIdx.x;
    int bl = blockIdx.y;
    int b  = b_start + bl;
    const float* Db = D + (size_t)bl * d_batch_stride;

    float bd[K_NN];
    int   bi[K_NN];
    #pragma unroll
    for (int j = 0; j < K_NN; ++j) { bd[j] = 3.402823e38f; bi[j] = 0; }

    for (int m = 0; m < N_PTS; ++m) {
        if (m == n) continue;
        float d = Db[(size_t)m * N_PTS + n];   // d[m][n] == d[n][m]
        if (d < bd[K_NN - 1]) {
            int p = K_NN - 1;
            while (p > 0 && d < bd[p - 1]) {
                bd[p] = bd[p - 1]; bi[p] = bi[p - 1]; --p;
            }
            bd[p] = d; bi[p] = m;
        }
    }

    int* ib = idxbuf + (size_t)bl * idx_batch_stride;
    #pragma unroll
    for (int j = 0; j < K_NN; ++j) {
        ib[(size_t)n * K_NN + j] = bi[j];
        // idx_out layout: (B, k, N), stored as float in d_out tail
        idx_out[((size_t)b * K_NN + j) * N_PTS + n] = (float)bi[j];
    }
}

// -------- edge feature: out[b, c, n, j] = central ; out[b, 64+c, n, j] = nb - central
// grid = (N*K/256, b_count), block = 256. Thread per (n, j). Writes are coalesced
// (j fastest, then n) for every c; reads of X hit L2 (X_b = 512 KB).
__global__ void __launch_bounds__(256)
gather_kernel(const float* __restrict__ x, const int* __restrict__ idxbuf,
              float* __restrict__ out, int b_start, unsigned long long idx_batch_stride) {
    int t  = blockIdx.x * 256 + threadIdx.x;
    int n  = t >> 4;
    int j  = t & 15;
    int bl = blockIdx.y;
    int b  = b_start + bl;

    int nb = idxbuf[(size_t)bl * idx_batch_stride + (size_t)n * K_NN + j];
    const float* X = x + (size_t)b * C_DIM * N_PTS;
    float* ob = out + (size_t)b * (2ull * C_DIM * N_PTS * K_NN);

    #pragma unroll 4
    for (int c = 0; c < C_DIM; ++c) {
        float ctr = X[c * N_PTS + n];
        float nv  = X[c * N_PTS + nb];
        ob[((size_t)c * N_PTS + n) * K_NN + j]           = ctr;
        ob[((size_t)(c + C_DIM) * N_PTS + n) * K_NN + j] = nv - ctr;
    }
}

extern "C" void kernel_launch(void* const* d_in, const int* in_sizes, int n_in,
                              void* d_out, int out_size, void* d_ws, size_t ws_size,
                              hipStream_t stream) {
    const float* x = (const float*)d_in[0];   // (8, 64, 2048) fp32
    // d_in[1] = k (scalar, always 16 here)
    float* out = (float*)d_out;
    const size_t edge_elems = (size_t)B_SZ * 2 * C_DIM * N_PTS * K_NN;  // 33,554,432
    float* idx_out = out + edge_elems;                                   // (B, k, N) as float

    const size_t sq_bytes   = (size_t)B_SZ * N_PTS * sizeof(float);           // 64 KB
    const size_t D_full     = (size_t)B_SZ * N_PTS * N_PTS * sizeof(float);   // 134 MB
    const size_t D_one      = (size_t)N_PTS * N_PTS * sizeof(float);          // 16.8 MB
    const size_t idx_full   = (size_t)B_SZ * N_PTS * K_NN * sizeof(int);      // 1 MB
    const size_t idx_one    = (size_t)N_PTS * K_NN * sizeof(int);             // 128 KB

    if (ws_size >= D_full + sq_bytes + idx_full) {
        // Single fused pass: D for all batches stays L2-resident (134 MB < 192 MB L2)
        float* Dbuf = (float*)d_ws;
        float* sqb  = (float*)((char*)d_ws + D_full);
        int*   ib   = (int*)((char*)d_ws + D_full + sq_bytes);

        norms_kernel<<<(B_SZ * N_PTS) / 256, 256, 0, stream>>>(x, sqb);
        dist_wmma_kernel<<<dim3(16, 128, B_SZ), 256, 0, stream>>>(
            x, sqb, Dbuf, 0, (unsigned long long)N_PTS * N_PTS);
        topk_kernel<<<dim3(N_PTS / 256, B_SZ), 256, 0, stream>>>(
            Dbuf, ib, idx_out, 0,
            (unsigned long long)N_PTS * N_PTS, (unsigned long long)N_PTS * K_NN);
        gather_kernel<<<dim3((N_PTS * K_NN) / 256, B_SZ), 256, 0, stream>>>(
            x, ib, out, 0, (unsigned long long)N_PTS * K_NN);
    } else {
        // Per-batch pipeline: reuse one 16.8 MB D buffer (stream-ordered, deterministic)
        float* Dbuf = (float*)d_ws;
        float* sqb  = (float*)((char*)d_ws + D_one);
        int*   ib   = (int*)((char*)d_ws + D_one + sq_bytes);

        norms_kernel<<<(B_SZ * N_PTS) / 256, 256, 0, stream>>>(x, sqb);
        for (int b = 0; b < B_SZ; ++b) {
            dist_wmma_kernel<<<dim3(16, 128, 1), 256, 0, stream>>>(x, sqb, Dbuf, b, 0ull);
            topk_kernel<<<dim3(N_PTS / 256, 1), 256, 0, stream>>>(
                Dbuf, ib, idx_out, b, 0ull, 0ull);
            gather_kernel<<<dim3((N_PTS * K_NN) / 256, 1), 256, 0, stream>>>(
                x, ib, out, b, 0ull);
        }
        (void)idx_one;
    }
    (void)in_sizes; (void)n_in; (void)out_size;
}